// LightconvLayer_8409545966487
// MI455X (gfx1250) — compile-verified
//
#include <hip/hip_runtime.h>

typedef float v2f __attribute__((ext_vector_type(2)));
typedef float v8f __attribute__((ext_vector_type(8)));

#define T_DIM   4096
#define B_DIM   8
#define C_DIM   1024
#define H_DIM   16
#define K_TAPS  31
#define BC      (B_DIM * C_DIM)      // 8192
#define NSTRIP  8
#define STRIP_T (T_DIM / NSTRIP)     // 512
#define NIT     (STRIP_T / 16)       // 32 windows of 16 outputs per strip

// out[t,b,c] = sum_k softmax(w)[h,k] * x[t+k-30, b, c]   (h = c/64)
// Tile: 16 timesteps x 16 channels = A(16x48 banded Toeplitz) * Xin(48x16),
// accumulated as 12 x V_WMMA_F32_16X16X4_F32. Xin slides in registers.
__global__ __launch_bounds__(256) void lightconv_wmma_kernel(
    const float* __restrict__ x, const float* __restrict__ wgt,
    float* __restrict__ out) {
  const int lane = threadIdx.x & 31;
  const int r    = lane & 15;   // matrix row (A/D) or column (B/D)
  const int half = lane >> 4;   // lane half selects K/M sub-rows per ISA layout

  // one wave per (b, channel-group, strip): 8 * 64 * 8 = 4096 waves
  const int gw    = blockIdx.x * 8 + (threadIdx.x >> 5);
  const int strip = gw & (NSTRIP - 1);
  const int cg    = (gw >> 3) & 63;       // 64 groups of 16 channels
  const int b     = gw >> 9;              // batch
  const int h     = cg >> 2;              // 4 channel-groups per head
  const int baseCh = b * C_DIM + cg * 16 + r;
  const int tstart = strip * STRIP_T;

  // ---- softmax of the 31 taps for this head (one-time per wave) ----
  const float* wh = wgt + h * K_TAPS;
  float mx = wh[0];
#pragma unroll
  for (int k = 1; k < K_TAPS; ++k) mx = fmaxf(mx, wh[k]);
  float s = 0.f;
#pragma unroll
  for (int k = 0; k < K_TAPS; ++k) s += __expf(wh[k] - mx);
  const float rs = 1.f / s;

  // ---- banded Toeplitz A: 16x48 split into 12 chunks of 16x4 ----
  // ISA 32-bit A 16x4 layout: lanes 0-15 row M, VGPR0/1 = K0,K1; lanes 16-31 = K2,K3
  v2f a[12];
#pragma unroll
  for (int kc = 0; kc < 12; ++kc) {
#pragma unroll
    for (int v = 0; v < 2; ++v) {
      const int col = 4 * kc + 2 * half + v;   // q index in window
      const int d   = col - r;                 // tap index
      a[kc][v] = (d >= 0 && d < K_TAPS) ? __expf(wh[d] - mx) * rs : 0.f;
    }
  }

  // ---- prologue: fill sliding window (48 timesteps x 16 channels) ----
  // chunk kc holds q = 4kc..4kc+3; element v at u = t0 - 30 + 4kc + 2*half + v
  v2f bx[12];
#pragma unroll
  for (int kc = 0; kc < 12; ++kc) {
#pragma unroll
    for (int v = 0; v < 2; ++v) {
      const int u = tstart - 30 + 4 * kc + 2 * half + v;
      bx[kc][v] = (u >= 0 && u < T_DIM) ? x[(long)u * BC + baseCh] : 0.f;
    }
  }

  for (int it = 0; it < NIT; ++it) {
    const int t0 = tstart + 16 * it;

    v8f acc = 0.f;
#pragma unroll
    for (int kc = 0; kc < 12; ++kc) {
      // 8-arg form: (neg_a, A, neg_b, B, c_mod, C, reuse_a, reuse_b)
      acc = __builtin_amdgcn_wmma_f32_16x16x4_f32(
          false, a[kc], false, bx[kc], (short)0, acc, false, false);
    }

    // D layout: VGPR v = row (v + 8*half), column r
#pragma unroll
    for (int v = 0; v < 8; ++v) {
      out[(long)(t0 + v + 8 * half) * BC + baseCh] = acc[v];
    }

    if (it + 1 < NIT) {
      // slide window: next t0' = t0+16 -> q' = q-16 -> chunk shift by 4
#pragma unroll
      for (int kc = 0; kc < 8; ++kc) bx[kc] = bx[kc + 4];
#pragma unroll
      for (int kc = 8; kc < 12; ++kc) {
#pragma unroll
        for (int v = 0; v < 2; ++v) {
          const int u = (t0 + 16) - 30 + 4 * kc + 2 * half + v;
          bx[kc][v] = (u >= 0 && u < T_DIM) ? x[(long)u * BC + baseCh] : 0.f;
        }
      }
    }
  }
}

extern "C" void kernel_launch(void* const* d_in, const int* in_sizes, int n_in,
                              void* d_out, int out_size, void* d_ws, size_t ws_size,
                              hipStream_t stream) {
  (void)in_sizes; (void)n_in; (void)out_size; (void)d_ws; (void)ws_size;
  const float* x   = (const float*)d_in[0];   // [T, B, C] fp32
  const float* wgt = (const float*)d_in[1];   // [H, K] fp32
  float* out = (float*)d_out;                 // [T, B, C] fp32

  const int total_waves = B_DIM * (C_DIM / 16) * NSTRIP;  // 4096
  dim3 grid(total_waves / 8);                              // 512 blocks x 8 waves
  dim3 block(256);
  lightconv_wmma_kernel<<<grid, block, 0, stream>>>(x, wgt, out);
}